// SGC_57647051047654
// MI455X (gfx1250) — compile-verified
//
#include <hip/hip_runtime.h>
#include <cstdint>

// SGC (K=2) for MI455X / gfx1250, wave32.
// Pipeline: degree -> norm -> (scale, scatter-add) x2 -> WMMA f32 GEMM.

#define TPB 256

typedef float v2f __attribute__((ext_vector_type(2)));
typedef float v8f __attribute__((ext_vector_type(8)));

// ---------------- utility kernels ----------------

__global__ void sgc_zero_kernel(float* __restrict__ p, long long n) {
    long long i = (long long)blockIdx.x * blockDim.x + threadIdx.x;
    if (i < n) p[i] = 0.0f;
}

__global__ void sgc_deg_kernel(const int* __restrict__ dst,
                               unsigned* __restrict__ deg, int nE) {
    int e = blockIdx.x * blockDim.x + threadIdx.x;
    if (e < nE) atomicAdd(&deg[dst[e]], 1u);
}

// in-place: buf holds uint degree, becomes float norm = rsqrt(max(deg,1))
__global__ void sgc_norm_kernel(float* __restrict__ buf, int nN) {
    int i = blockIdx.x * blockDim.x + threadIdx.x;
    if (i < nN) {
        unsigned d = ((const unsigned*)buf)[i];
        float df = (float)d;
        df = fmaxf(df, 1.0f);
        buf[i] = rsqrtf(df);
    }
}

// out[node,:] = in[node,:] * norm[node]^(1 or 2), vectorized float4
__global__ void sgc_scale_kernel(const float* __restrict__ in,
                                 const float* __restrict__ norm,
                                 float* __restrict__ out,
                                 int nN, int inF, int squared) {
    long long i = (long long)blockIdx.x * blockDim.x + threadIdx.x;
    long long total = (long long)nN * (inF / 4);
    if (i >= total) return;
    int node = (int)(i / (inF / 4));
    float s = norm[node];
    if (squared) s *= s;
    float4 v = ((const float4*)in)[i];
    v.x *= s; v.y *= s; v.z *= s; v.w *= s;
    ((float4*)out)[i] = v;
}

// one block (inF threads) per edge: hout[dst,:] += hin[src,:]
__global__ void sgc_prop_kernel(const int* __restrict__ src,
                                const int* __restrict__ dst,
                                const float* __restrict__ hin,
                                float* __restrict__ hout,
                                int inF) {
    int e = blockIdx.x;            // uniform -> scalar loads of indices
    int s = src[e];
    int d = dst[e];
    int f = threadIdx.x;
    float v = hin[(long long)s * inF + f];
    unsafeAtomicAdd(&hout[(long long)d * inF + f], v);  // global_atomic_add_f32
}

// ---------------- WMMA f32 GEMM: out[M,outF] = H[M,inF] @ W[inF,outF] + bias ----------------
// One wave per 16x16 C tile. K stepped by 4 via V_WMMA_F32_16X16X4_F32.
// A layout (16x4 f32): lanes 0-15 -> M=lane, VGPR0/1 = K{0,1}; lanes 16-31 -> K{2,3}.
// B layout (4x16 f32): lanes 0-15 -> N=lane, VGPR0/1 = K{0,1}; lanes 16-31 -> K{2,3}.
// C/D layout: VGPR v -> row m0 + v + 8*(lane>=16), col n0 + (lane&15).
__global__ void sgc_gemm_wmma(const float* __restrict__ H,
                              const float* __restrict__ Wm,
                              const float* __restrict__ bias,
                              float* __restrict__ out,
                              int M, int inF, int outF) {
    const int lane = threadIdx.x & 31;
    const int wave = threadIdx.x >> 5;     // which 16-col tile of outF
    const int m0 = blockIdx.x * 16;
    const int n0 = wave * 16;
    const int mr = lane & 15;
    const int hi = lane >> 4;              // 0 or 1
    const int klo = hi * 2;

    int row = m0 + mr;
    if (row >= M) row = M - 1;             // clamp: WMMA requires full EXEC
    const float* __restrict__ hrow = H + (long long)row * inF;

    v8f c = {};
    for (int k = 0; k < inF; k += 4) {
        v2f a, b;
        a.x = hrow[k + klo];
        a.y = hrow[k + klo + 1];
        const float* wp = Wm + (long long)(k + klo) * outF + n0 + mr;
        b.x = wp[0];
        b.y = wp[outF];
        c = __builtin_amdgcn_wmma_f32_16x16x4_f32(false, a, false, b,
                                                  (short)0, c, false, false);
    }

    const int n = n0 + mr;
    const float bb = bias[n];
    const int rbase = m0 + hi * 8;
#pragma unroll
    for (int v = 0; v < 8; ++v) {
        int r = rbase + v;
        if (r < M) out[(long long)r * outF + n] = c[v] + bb;
    }
}

// ---------------- launcher ----------------

extern "C" void kernel_launch(void* const* d_in, const int* in_sizes, int n_in,
                              void* d_out, int out_size, void* d_ws, size_t ws_size,
                              hipStream_t stream) {
    const float* x    = (const float*)d_in[0];   // [N, inF]
    const int*   src  = (const int*)d_in[1];     // [E]
    const int*   dst  = (const int*)d_in[2];     // [E]
    const float* Wm   = (const float*)d_in[3];   // [inF, outF]
    const float* bias = (const float*)d_in[4];   // [outF]
    float* out = (float*)d_out;                  // [N, outF]

    const int outF = in_sizes[4];
    const int inF  = in_sizes[3] / outF;
    const int nE   = in_sizes[1];
    const int nN   = in_sizes[0] / inF;

    // workspace layout: [norm (padded)] [bufA N*inF] [bufB N*inF]
    float* normb = (float*)d_ws;
    long long nodesPad = (((long long)nN + 255) / 256) * 256;  // float4-aligned
    float* bufA = normb + nodesPad;
    long long featN = (long long)nN * inF;
    float* bufB = bufA + featN;

    dim3 blk(TPB);

    // degree + norm
    sgc_zero_kernel<<<(unsigned)((nN + TPB - 1) / TPB), blk, 0, stream>>>(normb, nN);
    sgc_deg_kernel<<<(unsigned)((nE + TPB - 1) / TPB), blk, 0, stream>>>(dst, (unsigned*)normb, nE);
    sgc_norm_kernel<<<(unsigned)((nN + TPB - 1) / TPB), blk, 0, stream>>>(normb, nN);

    long long q = featN / 4;
    unsigned gq = (unsigned)((q + TPB - 1) / TPB);
    unsigned gf = (unsigned)((featN + TPB - 1) / TPB);

    // round 1: pre-scale, scatter
    sgc_scale_kernel<<<gq, blk, 0, stream>>>(x, normb, bufA, nN, inF, 0);
    sgc_zero_kernel<<<gf, blk, 0, stream>>>(bufB, featN);
    sgc_prop_kernel<<<(unsigned)nE, dim3(inF), 0, stream>>>(src, dst, bufA, bufB, inF);

    // round 1 post-scale fused with round 2 pre-scale (norm^2)
    sgc_scale_kernel<<<gq, blk, 0, stream>>>(bufB, normb, bufA, nN, inF, 1);
    sgc_zero_kernel<<<gf, blk, 0, stream>>>(bufB, featN);
    sgc_prop_kernel<<<(unsigned)nE, dim3(inF), 0, stream>>>(src, dst, bufA, bufB, inF);

    // final post-scale
    sgc_scale_kernel<<<gq, blk, 0, stream>>>(bufB, normb, bufA, nN, inF, 0);

    // projection via f32 WMMA
    sgc_gemm_wmma<<<(unsigned)((nN + 15) / 16), dim3((outF / 16) * 32), 0, stream>>>(
        bufA, Wm, bias, out, nN, inF, outF);
}